// MultiClassEdgeNetwork_37383395344660
// MI455X (gfx1250) — compile-verified
//
#include <hip/hip_runtime.h>
#include <hip/hip_bf16.h>
#include <math.h>

// MultiClassEdgeNetwork on gfx1250:
//   per edge e, per class c: h = tanh(B[e,c,:] @ W1[c] + b1[c]); o = tanh(h @ W2[c] + b2[c])
//   out[e,c] = softmax_c(o)
// Layer 1 mapped onto V_WMMA_F32_16X16X4_F32 (K=4 == in_feats, fp32):
//   A = W1[c]^T (rows = hid, padded to 16), B = per-edge features (cols = 16 edges),
//   C = b1 broadcast. Result: each lane (0-15) holds the full 8-wide h vector for
//   its edge -> layer 2 + softmax are pure per-lane, branch-free VALU
//   (hardware v_exp_f32 / v_rcp_f32, no libm branches between WMMAs).

typedef __attribute__((ext_vector_type(2))) float v2f;
typedef __attribute__((ext_vector_type(4))) float v4f;
typedef __attribute__((ext_vector_type(8))) float v8f;

#define NCLS 4
#define HID_ 8

// Branchless tanh: 1 - 2/(exp(2x)+1). exp(2x)->inf gives +1, ->0 gives -1.
__device__ __forceinline__ float fast_tanh(float x) {
    const float e = __expf(2.0f * x);                    // v_exp_f32
    return 1.0f - 2.0f * __builtin_amdgcn_rcpf(e + 1.0f); // v_rcp_f32
}

__global__ __launch_bounds__(256)
void edge_mlp_softmax_wmma(const float* __restrict__ m,          // [N, 4, 2]
                           const long long* __restrict__ ei,     // [2, E]
                           const float* __restrict__ W1,         // [4, 4, 8]
                           const float* __restrict__ b1,         // [4, 8]
                           const float* __restrict__ W2,         // [4, 8, 1]
                           const float* __restrict__ b2,         // [4, 1]
                           float* __restrict__ out,              // [E, 4]
                           long long E)
{
    const int lane = threadIdx.x & 31;
    const int wave = threadIdx.x >> 5;
    const long long tile = (long long)blockIdx.x * 8 + wave;   // 16 edges per wave
    const long long e0   = tile * 16;
    if (e0 >= E) return;                                       // wave-uniform exit

    const int n    = lane & 15;   // edge-in-tile (B/D column) == A row (hid)
    const int half = lane >> 4;   // 0: K={0,1} -> dst node (col); 1: K={2,3} -> src node (row)
    const int k0   = half * 2;

    long long e = e0 + n;
    const long long eld = (e < E) ? e : (E - 1);               // clamp loads, keep EXEC full

    // edge_index[0] = row (src), edge_index[1] = col (dst); concat order is [m[col], m[row]]
    const long long node  = ei[half ? eld : (E + eld)];
    const float*    mbase = m + node * (NCLS * 2);

    const int rr = n & 7;                // safe hid row for W1 loads
    const bool arow_valid = (n < HID_);  // A rows 8..15 are zero padding

    float o[NCLS];

    #pragma unroll
    for (int c = 0; c < NCLS; ++c) {
        // ---- A-matrix: W1[c]^T, lane holds row n, K = k0..k0+1 (branchless zero pad)
        const float* w1c = W1 + c * (4 * HID_);   // W1[c][k][h]
        const float ax = w1c[(k0 + 0) * HID_ + rr];
        const float ay = w1c[(k0 + 1) * HID_ + rr];
        v2f a;
        a.x = arow_valid ? ax : 0.0f;
        a.y = arow_valid ? ay : 0.0f;

        // ---- B-matrix: column n = edge features; lane loads m[node, c, 0..1] (8B gather)
        v2f b;
        b.x = mbase[c * 2 + 0];
        b.y = mbase[c * 2 + 1];

        // ---- C init: b1[c][row] for real rows (lanes 0-15), zero for padded rows 8-15
        v8f acc;
        #pragma unroll
        for (int r = 0; r < HID_; ++r) {
            const float bv = b1[c * HID_ + r];
            acc[r] = half ? 0.0f : bv;
        }

        // D[hid][edge] = sum_k W1[c][k][hid] * feat[k][edge] + b1[c][hid]
        acc = __builtin_amdgcn_wmma_f32_16x16x4_f32(
                /*neg_a=*/false, a, /*neg_b=*/false, b,
                /*c_mod=*/(short)0, acc, /*reuse_a=*/false, /*reuse_b=*/false);

        // ---- layer 2: per-lane 8-wide tanh-dot (lane n holds h[hid r] in acc[r])
        float s = b2[c];
        #pragma unroll
        for (int r = 0; r < HID_; ++r)
            s += fast_tanh(acc[r]) * W2[c * HID_ + r];
        o[c] = fast_tanh(s);
    }

    // ---- softmax over the 4 classes (per lane, branch-free)
    const float mx = fmaxf(fmaxf(o[0], o[1]), fmaxf(o[2], o[3]));
    float ex[NCLS];
    float sum = 0.0f;
    #pragma unroll
    for (int c = 0; c < NCLS; ++c) { ex[c] = __expf(o[c] - mx); sum += ex[c]; }
    const float inv = __builtin_amdgcn_rcpf(sum);

    // ---- store: one float4 per edge (predicated; all WMMAs already done)
    if (half == 0 && e < E) {
        v4f r4;
        r4.x = ex[0] * inv;
        r4.y = ex[1] * inv;
        r4.z = ex[2] * inv;
        r4.w = ex[3] * inv;
        *(v4f*)(out + e * NCLS) = r4;
    }
}

extern "C" void kernel_launch(void* const* d_in, const int* in_sizes, int n_in,
                              void* d_out, int out_size, void* d_ws, size_t ws_size,
                              hipStream_t stream) {
    const float*     m  = (const float*)d_in[0];      // [N,4,2] f32
    const long long* ei = (const long long*)d_in[1];  // [2,E] i64
    const float*     W1 = (const float*)d_in[2];      // [4,4,8]
    const float*     b1 = (const float*)d_in[3];      // [4,8]
    const float*     W2 = (const float*)d_in[4];      // [4,8,1]
    const float*     b2 = (const float*)d_in[5];      // [4,1]
    float*           out = (float*)d_out;             // [E,4,1] f32

    const long long E = (long long)in_sizes[1] / 2;

    // 256 threads = 8 waves = 128 edges per block
    const long long blocks = (E + 127) / 128;
    edge_mlp_softmax_wmma<<<(int)blocks, 256, 0, stream>>>(m, ei, W1, b1, W2, b2, out, E);
}